// Encoder_16252156248330
// MI455X (gfx1250) — compile-verified
//
#include <hip/hip_runtime.h>

typedef float v2f __attribute__((ext_vector_type(2)));
typedef float v8f __attribute__((ext_vector_type(8)));

#define GRU_H 16
#define GRU_S 1024
#define GRU_B 1024

#define L2E 1.44269504088896340736f  // log2(e)

static __device__ __forceinline__ v8f wmma_f32_16x16x4(v2f a, v2f b, v8f c) {
  // D = A(16x4 f32) * B(4x16 f32) + C(16x16 f32)
  return __builtin_amdgcn_wmma_f32_16x16x4_f32(false, a, false, b, (short)0, c,
                                               false, false);
}

// One wave (32 threads) per (16-row batch tile, direction).
// h state lives in WMMA C/D layout: lane holds element (m = j + 8*(lane>=16),
// n = lane&15) in component j. Each step h is transposed via LDS into the
// 16x4-chunk A layout to feed the recurrent WMMA chains.
__global__ __launch_bounds__(32)
void gru_bidir_kernel(const float* __restrict__ x,       // (B, S, 4)
                      const float* __restrict__ w_ih_f,  // (48, 4)
                      const float* __restrict__ w_hh_f,  // (48, 16)
                      const float* __restrict__ b_ih_f,  // (48)
                      const float* __restrict__ b_hh_f,  // (48)
                      const float* __restrict__ w_ih_b,
                      const float* __restrict__ w_hh_b,
                      const float* __restrict__ b_ih_b,
                      const float* __restrict__ b_hh_b,
                      float* __restrict__ out,           // (B, S, 32)
                      float* __restrict__ hcat)          // (B, 32)
{
  const int lane = threadIdx.x;   // 0..31
  const int half = lane >> 4;     // lane-half selects K pairs / M rows 8..15
  const int lm   = lane & 15;
  const int tile = blockIdx.x;    // 0..63 : batch tile of 16 rows
  const int dir  = blockIdx.y;    // 0 = forward, 1 = backward

  const float* w_ih = dir ? w_ih_b : w_ih_f;
  const float* w_hh = dir ? w_hh_b : w_hh_f;
  const float* b_ih = dir ? b_ih_b : b_ih_f;
  const float* b_hh = dir ? b_hh_b : b_hh_f;

  // B fragments of w_hh^T (K=16 hidden x N=48 gates): 3 gate tiles x 4 K-chunks.
  // B layout (4x16): vgpr0 -> K = 4c + 2*half, vgpr1 -> K+1, N = 16g + lm.
  v2f whh[3][4];
#pragma unroll
  for (int g = 0; g < 3; ++g)
#pragma unroll
    for (int c = 0; c < 4; ++c)
      whh[g][c] = *(const v2f*)(w_hh + (16 * g + lm) * GRU_H + 4 * c + 2 * half);

  // B fragments of w_ih^T (K=4 x N=48): 3 gate tiles, single K chunk.
  v2f wih[3];
#pragma unroll
  for (int g = 0; g < 3; ++g)
    wih[g] = *(const v2f*)(w_ih + (16 * g + lm) * 4 + 2 * half);

  // Biases folded into per-lane activation constants (value depends on n = lm
  // only). r/z: sigmoid(t+b) = rcp(1 + exp2(t*(-L2E) + (-b*L2E))).
  // n: tanh(xg + bi + r*(hg + bh)) = 2*rcp(1 + exp2(t2*(-2L2E) + (-bi*2L2E))) - 1.
  const float bi0 = b_ih[lm],      bh0 = b_hh[lm];
  const float bi1 = b_ih[16 + lm], bh1 = b_hh[16 + lm];
  const float bi2 = b_ih[32 + lm], bh2 = b_hh[32 + lm];
  const float cr = -(bi0 + bh0) * L2E;
  const float cz = -(bi1 + bh1) * L2E;
  const float cn = -bi2 * (2.0f * L2E);

  v8f h = {};        // h0 = 0, C/D layout
  const v8f ZERO{};  // inline-0 WMMA C operand (no register copies)

  // Padded so the two lane-halves of the A-layout b64 reads hit disjoint banks.
  __shared__ __align__(16) float lds[16][20];

  const int b0 = tile * 16;
  const int brow = b0 + lm;
  const int s0 = dir ? (GRU_S - 1) : 0;
  const int sstep = dir ? -1 : 1;

  // Running pointers: per-step address math reduces to two pointer adds;
  // the 8 output stores use compile-time immediate offsets (j*128KB < 2^23).
  const float* xp = x + ((size_t)brow * GRU_S + s0) * 4 + 2 * half;
  float* op = out + ((size_t)(b0 + 8 * half) * GRU_S + s0) * 32 + dir * 16 + lm;

  for (int t = 0; t < GRU_S; ++t) {
    // x A-fragment (16x4): lane half 0 reads x[b][s][0:2], half 1 reads [2:4].
    v2f xa = *(const v2f*)xp;

    // xg = x @ w_ih^T   (3 WMMAs, inline-0 C)
    v8f xg0 = wmma_f32_16x16x4(xa, wih[0], ZERO);
    v8f xg1 = wmma_f32_16x16x4(xa, wih[1], ZERO);
    v8f xg2 = wmma_f32_16x16x4(xa, wih[2], ZERO);

    // Transpose h: D layout (vgpr=m&7, half=m>>3, lane=n) -> A layout
    // (lane=m, vgpr/half pick K) through LDS.
#pragma unroll
    for (int v = 0; v < 8; ++v) lds[v + 8 * half][lm] = h[v];
    __syncthreads();  // single-wave WG: barrier degenerates to the dscnt wait
    v2f ha[4];
#pragma unroll
    for (int c = 0; c < 4; ++c)
      ha[c] = *(const v2f*)(&lds[lm][4 * c + 2 * half]);
    __syncthreads();

    // hg = h @ w_hh^T : per gate, two independent 2-deep WMMA chains
    // (halves the dependent-WMMA depth on the step critical path).
    v8f a0 = wmma_f32_16x16x4(ha[0], whh[0][0], ZERO);
    v8f a1 = wmma_f32_16x16x4(ha[0], whh[1][0], ZERO);
    v8f a2 = wmma_f32_16x16x4(ha[0], whh[2][0], ZERO);
    v8f b0v = wmma_f32_16x16x4(ha[2], whh[0][2], ZERO);
    v8f b1v = wmma_f32_16x16x4(ha[2], whh[1][2], ZERO);
    v8f b2v = wmma_f32_16x16x4(ha[2], whh[2][2], ZERO);
    a0 = wmma_f32_16x16x4(ha[1], whh[0][1], a0);
    a1 = wmma_f32_16x16x4(ha[1], whh[1][1], a1);
    a2 = wmma_f32_16x16x4(ha[1], whh[2][1], a2);
    b0v = wmma_f32_16x16x4(ha[3], whh[0][3], b0v);
    b1v = wmma_f32_16x16x4(ha[3], whh[1][3], b1v);
    b2v = wmma_f32_16x16x4(ha[3], whh[2][3], b2v);
    v8f hg0 = a0 + b0v;
    v8f hg1 = a1 + b1v;
    v8f hg2 = a2 + b2v;

    // GRU gate math, elementwise in C/D layout; biases folded into fma consts.
    v8f hn;
#pragma unroll
    for (int j = 0; j < 8; ++j) {
      float t0 = xg0[j] + hg0[j];
      float r = __builtin_amdgcn_rcpf(
          1.0f + __builtin_amdgcn_exp2f(__builtin_fmaf(t0, -L2E, cr)));
      float t1 = xg1[j] + hg1[j];
      float z = __builtin_amdgcn_rcpf(
          1.0f + __builtin_amdgcn_exp2f(__builtin_fmaf(t1, -L2E, cz)));
      float hg2b = hg2[j] + bh2;
      float t2 = __builtin_fmaf(r, hg2b, xg2[j]);
      float e = __builtin_amdgcn_exp2f(__builtin_fmaf(t2, -2.0f * L2E, cn));
      float nn = __builtin_fmaf(2.0f, __builtin_amdgcn_rcpf(1.0f + e), -1.0f);
      hn[j] = nn + z * (h[j] - nn);   // == (1-z)*n + z*h
    }
    h = hn;

    // out[b][s][dir*16 + n], b = b0 + j + 8*half, n = lm
#pragma unroll
    for (int j = 0; j < 8; ++j)
      op[(size_t)j * GRU_S * 32] = h[j];

    xp += 4 * sstep;
    op += 32 * sstep;
  }

  // Final hidden state: hcat[b][dir*16 + n]
#pragma unroll
  for (int j = 0; j < 8; ++j)
    hcat[(size_t)(b0 + j + 8 * half) * 32 + dir * 16 + lm] = h[j];
}

extern "C" void kernel_launch(void* const* d_in, const int* in_sizes, int n_in,
                              void* d_out, int out_size, void* d_ws, size_t ws_size,
                              hipStream_t stream) {
  (void)in_sizes; (void)n_in; (void)out_size; (void)d_ws; (void)ws_size;
  const float* x      = (const float*)d_in[0];
  const float* w_ih_f = (const float*)d_in[1];
  const float* w_hh_f = (const float*)d_in[2];
  const float* b_ih_f = (const float*)d_in[3];
  const float* b_hh_f = (const float*)d_in[4];
  const float* w_ih_b = (const float*)d_in[5];
  const float* w_hh_b = (const float*)d_in[6];
  const float* b_ih_b = (const float*)d_in[7];
  const float* b_hh_b = (const float*)d_in[8];

  float* out  = (float*)d_out;
  float* hcat = out + (size_t)GRU_B * GRU_S * 32;  // outputs concatenated flat

  dim3 grid(GRU_B / 16, 2);  // 64 batch tiles x 2 directions, 1 wave each
  dim3 block(32);
  gru_bidir_kernel<<<grid, block, 0, stream>>>(x, w_ih_f, w_hh_f, b_ih_f, b_hh_f,
                                               w_ih_b, w_hh_b, b_ih_b, b_hh_b,
                                               out, hcat);
}